// DynamicRouting_64647847739717
// MI455X (gfx1250) — compile-verified
//
#include <hip/hip_runtime.h>

typedef float v2f __attribute__((ext_vector_type(2)));
typedef float v8f __attribute__((ext_vector_type(8)));

#define B_SZ   16
#define C_CH   512
#define HW     196
#define N_PB   1568              // (512/64)*14*14 per batch
#define NT     25088             // B_SZ * N_PB
#define O_NUM  32
#define COS_EPS 1e-8f
#define SUM_EPS 1e-8f
#define LN_EPS  1e-5f

// workspace layout (float offsets)
#define OFF_SCALE 0                         // 32*64
#define OFF_U     2048                      // NT*64
#define OFF_UT    (OFF_U + NT*64)           // NT*64
#define OFF_UN    (OFF_UT + NT*64)          // NT
#define OFF_B     (OFF_UN + NT)             // NT*32 routing logits
#define OFF_M     (OFF_B + NT*O_NUM)        // B*O*64
#define OFF_CSUM  (OFF_M + B_SZ*O_NUM*64)   // B*O
#define OFF_S     (OFF_CSUM + B_SZ*O_NUM)   // B*O*64
#define OFF_V0    (OFF_S + B_SZ*O_NUM*64)   // B*O*64
#define OFF_MBAR  (OFF_V0 + B_SZ*O_NUM*64)  // B*64
#define OFF_VN    (OFF_MBAR + B_SZ*64)      // B*O

__device__ __forceinline__ v8f wmma4(v2f a, v2f b, v8f c) {
  return __builtin_amdgcn_wmma_f32_16x16x4_f32(false, a, false, b, (short)0, c,
                                               false, false);
}

// scale = LayerNorm(codes @ scale_weight^T) over in-dim; one block per o-row
__global__ void k_scale(const float* __restrict__ codes,
                        const float* __restrict__ sw, float* ws) {
  __shared__ float r1[64], r2[64];
  int o = blockIdx.x, i = threadIdx.x;
  float acc = 0.f;
  for (int d = 0; d < 64; ++d) acc += codes[o * 64 + d] * sw[i * 64 + d];
  r1[i] = acc; r2[i] = acc * acc;
  __syncthreads();
  for (int st = 32; st > 0; st >>= 1) {
    if (i < st) { r1[i] += r1[i + st]; r2[i] += r2[i + st]; }
    __syncthreads();
  }
  float m = r1[0] * (1.f / 64.f);
  float v = r2[0] * (1.f / 64.f) - m * m;
  ws[OFF_SCALE + o * 64 + i] = (acc - m) * rsqrtf(v + LN_EPS);
}

// materialize u (B,N,64) from feature (B,512,14,14)
__global__ void k_reshape(const float* __restrict__ feat, float* ws) {
  int idx = blockIdx.x * 256 + threadIdx.x;       // < NT*64
  int i = idx & 63;
  int rest = idx >> 6;
  int n = rest % N_PB;
  int b = rest / N_PB;
  int g = n / HW, hw = n % HW;
  ws[OFF_U + idx] = feat[(b * C_CH + g * 64 + i) * HW + hw];
}

// u_t = u @ transform_w^T + transform_b via f32 WMMA (16x16x4, K=64)
__global__ void k_ut(const float* __restrict__ tw, const float* __restrict__ tb,
                     float* ws) {
  __shared__ float ldsW[64 * 64];
  int t = threadIdx.x;
#pragma unroll
  for (int r = 0; r < 16; ++r) ldsW[r * 256 + t] = tw[r * 256 + t];
  __syncthreads();
  int wave = t >> 5, lane = t & 31;
  int row = lane & 15, hi = lane >> 4;
  int r0 = (blockIdx.x * 8 + wave) * 16;
  const float* uRow = ws + OFF_U + (size_t)(r0 + row) * 64;
  v8f a0 = {}, a1 = {}, a2 = {}, a3 = {};
#pragma unroll
  for (int k = 0; k < 16; ++k) {
    int kb = k * 4 + hi * 2;
    v2f a; a.x = uRow[kb]; a.y = uRow[kb + 1];
    v2f b0, b1, b2, b3;
    b0.x = ldsW[row * 64 + kb];        b0.y = ldsW[row * 64 + kb + 1];
    b1.x = ldsW[(16 + row) * 64 + kb]; b1.y = ldsW[(16 + row) * 64 + kb + 1];
    b2.x = ldsW[(32 + row) * 64 + kb]; b2.y = ldsW[(32 + row) * 64 + kb + 1];
    b3.x = ldsW[(48 + row) * 64 + kb]; b3.y = ldsW[(48 + row) * 64 + kb + 1];
    a0 = wmma4(a, b0, a0);
    a1 = wmma4(a, b1, a1);
    a2 = wmma4(a, b2, a2);
    a3 = wmma4(a, b3, a3);
  }
  float* ut = ws + OFF_UT;
#pragma unroll
  for (int r = 0; r < 8; ++r) {
    size_t gr = (size_t)(r0 + r + hi * 8) * 64;
    ut[gr + row]      = a0[r] + tb[row];
    ut[gr + 16 + row] = a1[r] + tb[16 + row];
    ut[gr + 32 + row] = a2[r] + tb[32 + row];
    ut[gr + 48 + row] = a3[r] + tb[48 + row];
  }
}

__global__ void k_un(float* ws) {
  int n = blockIdx.x * 256 + threadIdx.x;
  if (n >= NT) return;
  const float* r = ws + OFF_UT + (size_t)n * 64;
  float s = 0.f;
  for (int j = 0; j < 64; ++j) s += r[j] * r[j];
  ws[OFF_UN + n] = sqrtf(s);
}

__global__ void k_mbar(float* ws) {
  int b = blockIdx.x, i = threadIdx.x;
  const float* u = ws + OFF_U + (size_t)b * N_PB * 64;
  float s = 0.f;
  for (int n = 0; n < N_PB; ++n) s += u[n * 64 + i];
  ws[OFF_MBAR + b * 64 + i] = s * (1.f / (float)N_PB);
}

// v0[b,o,:] = W @ (scale[o] * Mbar[b]) + bias ; also init s = v0
__global__ void k_v0(const float* __restrict__ nw, const float* __restrict__ nb,
                     float* ws) {
  int bo = blockIdx.x, b = bo >> 5, o = bo & 31, j = threadIdx.x;
  const float* sc = ws + OFF_SCALE + o * 64;
  const float* mb = ws + OFF_MBAR + b * 64;
  float acc = 0.f;
  for (int i = 0; i < 64; ++i) acc += nw[j * 64 + i] * sc[i] * mb[i];
  float v0 = acc + nb[j];
  ws[OFF_V0 + bo * 64 + j] = v0;
  ws[OFF_S + bo * 64 + j] = v0;
}

__global__ void k_zero(float* p, int n) {
  int i = blockIdx.x * 256 + threadIdx.x;
  if (i < n) p[i] = 0.f;
}

__global__ void k_vn(float* ws) {
  int t = blockIdx.x * 256 + threadIdx.x;
  if (t >= B_SZ * O_NUM) return;
  const float* s = ws + OFF_S + t * 64;
  float a = 0.f;
  for (int j = 0; j < 64; ++j) a += s[j] * s[j];
  ws[OFF_VN + t] = sqrtf(a);
}

// sim = u_t @ s^T (WMMA), then b += sim/max(|u||s|,eps), c = softmax_o(b)
__global__ void k_sim(float* ws, float* __restrict__ c_out) {
  __shared__ float ldsS[O_NUM * 64];       // s for this batch
  __shared__ float simb[8 * 16 * O_NUM];   // per-wave 16x32 sim tile
  int b = blockIdx.y, t = threadIdx.x;
  const float* sB = ws + OFF_S + (size_t)b * O_NUM * 64;
#pragma unroll
  for (int r = 0; r < 8; ++r) ldsS[r * 256 + t] = sB[r * 256 + t];
  __syncthreads();
  int wave = t >> 5, lane = t & 31, row = lane & 15, hi = lane >> 4;
  int tile = blockIdx.x * 8 + wave;
  if (tile < 98) {
    const float* utRow = ws + OFF_UT + (size_t)(b * N_PB + tile * 16 + row) * 64;
    v8f a0 = {}, a1 = {};
#pragma unroll
    for (int k = 0; k < 16; ++k) {
      int kb = k * 4 + hi * 2;
      v2f a; a.x = utRow[kb]; a.y = utRow[kb + 1];
      v2f b0, b1;
      b0.x = ldsS[row * 64 + kb];        b0.y = ldsS[row * 64 + kb + 1];
      b1.x = ldsS[(16 + row) * 64 + kb]; b1.y = ldsS[(16 + row) * 64 + kb + 1];
      a0 = wmma4(a, b0, a0);
      a1 = wmma4(a, b1, a1);
    }
#pragma unroll
    for (int r = 0; r < 8; ++r) {
      int rl = r + hi * 8;
      simb[(wave * 16 + rl) * 32 + row] = a0[r];
      simb[(wave * 16 + rl) * 32 + 16 + row] = a1[r];
    }
  }
  __syncthreads();
  if (t < 128) {
    int w = t >> 4, rl = t & 15;
    int tib = blockIdx.x * 8 + w;
    if (tib < 98) {
      int gn = b * N_PB + tib * 16 + rl;
      float unv = ws[OFF_UN + gn];
      float* bl = ws + OFF_B + (size_t)gn * 32;
      const float* vn = ws + OFF_VN + b * 32;
      float bb[32];
      float mx = -1e30f;
#pragma unroll
      for (int o = 0; o < 32; ++o) {
        float sim = simb[(w * 16 + rl) * 32 + o] / fmaxf(unv * vn[o], COS_EPS);
        float nbv = bl[o] + sim;
        bl[o] = nbv;
        bb[o] = nbv;
        mx = fmaxf(mx, nbv);
      }
      float sum = 0.f;
#pragma unroll
      for (int o = 0; o < 32; ++o) { bb[o] = __expf(bb[o] - mx); sum += bb[o]; }
      float inv = 1.f / sum;
#pragma unroll
      for (int o = 0; o < 32; ++o) c_out[(size_t)gn * 32 + o] = bb[o] * inv;
    }
  }
}

// M[b,o,i] = sum_n c*u ; Csum[b,o] = sum_n c
__global__ void k_M(const float* __restrict__ c_out, float* ws) {
  int bo = blockIdx.x, b = bo >> 5, o = bo & 31, i = threadIdx.x;
  const float* u = ws + OFF_U + (size_t)b * N_PB * 64;
  const float* c = c_out + (size_t)b * N_PB * 32 + o;
  float acc = 0.f, cs = 0.f;
  for (int n = 0; n < N_PB; ++n) {
    float cv = c[(size_t)n * 32];
    acc += cv * u[n * 64 + i];
    cs += cv;
  }
  ws[OFF_M + bo * 64 + i] = acc;
  if (i == 0) ws[OFF_CSUM + bo] = cs;
}

// s[b,o,:] = (W @ (scale[o]*M[b,o]) + bias*Csum) / (eps + Csum)
__global__ void k_supd(const float* __restrict__ nw, const float* __restrict__ nb,
                       float* ws) {
  int bo = blockIdx.x, o = bo & 31, j = threadIdx.x;
  const float* sc = ws + OFF_SCALE + o * 64;
  const float* M = ws + OFF_M + bo * 64;
  float cs = ws[OFF_CSUM + bo];
  float acc = 0.f;
  for (int i = 0; i < 64; ++i) acc += nw[j * 64 + i] * sc[i] * M[i];
  ws[OFF_S + bo * 64 + j] = (acc + nb[j] * cs) / (SUM_EPS + cs);
}

__global__ void k_v(float* ws, float* __restrict__ out) {
  int idx = blockIdx.x * 256 + threadIdx.x;
  if (idx < B_SZ * O_NUM * 64) {
    float v = ws[OFF_V0 + idx] + ws[OFF_S + idx];
    out[idx] = v > 0.f ? v : 0.f;
  }
}

extern "C" void kernel_launch(void* const* d_in, const int* in_sizes, int n_in,
                              void* d_out, int out_size, void* d_ws, size_t ws_size,
                              hipStream_t stream) {
  (void)in_sizes; (void)n_in; (void)out_size; (void)ws_size;
  const float* feature   = (const float*)d_in[0];
  const float* native_w  = (const float*)d_in[1];
  const float* native_b  = (const float*)d_in[2];
  const float* scale_w   = (const float*)d_in[3];
  const float* codes     = (const float*)d_in[4];
  const float* transf_w  = (const float*)d_in[5];
  const float* transf_b  = (const float*)d_in[6];
  float* out = (float*)d_out;
  float* c_out = out + B_SZ * O_NUM * 64;   // v first, then c
  float* ws = (float*)d_ws;

  k_scale<<<O_NUM, 64, 0, stream>>>(codes, scale_w, ws);
  k_reshape<<<NT * 64 / 256, 256, 0, stream>>>(feature, ws);
  k_ut<<<NT / 16 / 8, 256, 0, stream>>>(transf_w, transf_b, ws);
  k_un<<<(NT + 255) / 256, 256, 0, stream>>>(ws);
  k_mbar<<<B_SZ, 64, 0, stream>>>(ws);
  k_v0<<<B_SZ * O_NUM, 64, 0, stream>>>(native_w, native_b, ws);
  k_zero<<<(NT * O_NUM + 255) / 256, 256, 0, stream>>>(ws + OFF_B, NT * O_NUM);

  for (int it = 0; it < 3; ++it) {
    k_vn<<<2, 256, 0, stream>>>(ws);
    k_sim<<<dim3(13, B_SZ), 256, 0, stream>>>(ws, c_out);
    k_M<<<B_SZ * O_NUM, 64, 0, stream>>>(c_out, ws);
    k_supd<<<B_SZ * O_NUM, 64, 0, stream>>>(native_w, native_b, ws);
  }
  k_v<<<(B_SZ * O_NUM * 64 + 255) / 256, 256, 0, stream>>>(ws, out);
}